// MPSClassifier_77412490543200
// MI455X (gfx1250) — compile-verified
//
#include <hip/hip_runtime.h>
#include <math.h>

// MPS classifier forward for MI455X (gfx1250, wave32).
// Tree contraction of 256-matrix chain, 128x128 f32 GEMMs via
// V_WMMA_F32_16X16X4_F32, per-node Frobenius normalization.
//
// Right-hand (B) operands are stored K-pair-interleaved:
//   elem(k, col) -> buf[(k>>1)*256 + col*2 + (k&1)]
// so each WMMA B-fragment (B[k][col], B[k+1][col]) is one aligned b64 LDS load.

typedef __attribute__((ext_vector_type(2))) float v2f;
typedef __attribute__((ext_vector_type(8))) float v8f;

#define BOND 128
#define NSITE 256
#define BATCH 64
#define MAT_ELEMS (BOND * BOND)       // 16384 floats per matrix
#define EPS 1e-12f

// ---------------------------------------------------------------------------
// Core: 128x128x128 f32 matmul, sA row-major, sB K-pair-interleaved.
// 256 threads = 8 waves; wave w owns output rows [16w,16w+16) = 8 tiles.
// Output stored row-major or interleaved depending on consumer side.
// ---------------------------------------------------------------------------
__device__ __forceinline__ void mm128_norm_store(const float* __restrict__ sA,
                                                 const float* __restrict__ sB,
                                                 float* __restrict__ outM,
                                                 float* __restrict__ outLog,
                                                 float* __restrict__ sred,
                                                 int storeInterleaved) {
    const int lane = threadIdx.x & 31;
    const int wave = threadIdx.x >> 5;    // 0..7
    const int half = lane >> 4;           // 0/1 (lane groups 0-15 / 16-31)
    const int l16  = lane & 15;

    v8f acc[8] = {};                      // 8 tiles x 8 VGPRs of f32 accum

    const int rowA = wave * 16 + l16;     // A-fragment row for this lane
    for (int k = 0; k < BOND; k += 4) {
        // A 16x4 fragment: lanes 0-15 hold K=k,k+1; lanes 16-31 hold K=k+2,k+3
        const v2f a = *(const v2f*)(sA + rowA * BOND + k + half * 2);
        // B fragments: interleaved layout -> contiguous aligned pair.
        const float* bbase = sB + k * 128 + half * 256;
#pragma unroll
        for (int j = 0; j < 8; ++j) {
            const int colB = j * 16 + l16;
            const v2f bfrag = *(const v2f*)(bbase + colB * 2);
            acc[j] = __builtin_amdgcn_wmma_f32_16x16x4_f32(
                false, a, false, bfrag, (short)0, acc[j], false, false);
        }
    }

    // Frobenius norm: per-lane sum of squares, fixed-order LDS tree reduction.
    float ss = 0.f;
#pragma unroll
    for (int j = 0; j < 8; ++j)
#pragma unroll
        for (int v = 0; v < 8; ++v) {
            const float e = acc[j][v];
            ss += e * e;
        }
    sred[threadIdx.x] = ss;
    __syncthreads();
    for (int s = 128; s > 0; s >>= 1) {
        if ((int)threadIdx.x < s) sred[threadIdx.x] += sred[threadIdx.x + s];
        __syncthreads();
    }
    float norm = sqrtf(sred[0]);
    norm = fmaxf(norm, EPS);
    if (threadIdx.x == 0) *outLog = logf(norm);
    const float inv = 1.0f / norm;

    // Store normalized tile: C/D layout -> VGPR v, lanes 0-15: M=v, 16-31: M=v+8.
#pragma unroll
    for (int j = 0; j < 8; ++j) {
        const int col = j * 16 + l16;
#pragma unroll
        for (int v = 0; v < 8; ++v) {
            const int row = wave * 16 + v + half * 8;
            const int idx = storeInterleaved
                                ? ((row >> 1) * 256 + col * 2 + (row & 1))
                                : (row * BOND + col);
            outM[idx] = acc[j][v] * inv;
        }
    }
}

// ---------------------------------------------------------------------------
// Level 0 site-matrix builders (fused; never materializes the 1 GB "others").
// Chain index m: 0 = padded special (row0 only), m>=1 uses cores[m-1], x[b][m].
// Left operands (even m) built row-major; right operands (odd m) interleaved.
// ---------------------------------------------------------------------------
__device__ __forceinline__ void build_site_rowmajor(float* __restrict__ dst,
                                                    const float* __restrict__ core0,
                                                    const float* __restrict__ cores,
                                                    const float* __restrict__ x,
                                                    int b, int m) {
    const int t = threadIdx.x;            // 0..255
    if (m == 0) {
        const float x0 = x[b * NSITE + 0];
        const float f0 = 1.f - x0, f1 = x0;
        for (int i = t; i < MAT_ELEMS; i += 256) dst[i] = 0.f;
        if (t < BOND) dst[t] = f0 * core0[t] + f1 * core0[BOND + t];  // row 0
    } else {
        const float xv = x[b * NSITE + m];
        const float f0 = 1.f - xv, f1 = xv;
        const float* cb = cores + (size_t)(m - 1) * (BOND * 2 * BOND);
        for (int i = t * 4; i < MAT_ELEMS; i += 256 * 4) {
            const int l = i >> 7;         // row
            const int h = i & 127;        // col
            const float* rp = cb + l * 256 + h;   // cores[n][l][0][h..]
            const float4 a0 = *(const float4*)(rp);
            const float4 a1 = *(const float4*)(rp + BOND);  // d=1 plane
            float4 r;
            r.x = f0 * a0.x + f1 * a1.x;
            r.y = f0 * a0.y + f1 * a1.y;
            r.z = f0 * a0.z + f1 * a1.z;
            r.w = f0 * a0.w + f1 * a1.w;
            *(float4*)(dst + i) = r;
        }
    }
}

__device__ __forceinline__ void build_site_interleaved(float* __restrict__ dst,
                                                       const float* __restrict__ cores,
                                                       const float* __restrict__ x,
                                                       int b, int m) {
    const int t = threadIdx.x;            // 0..255 ; m >= 1 always (odd index)
    const float xv = x[b * NSITE + m];
    const float f0 = 1.f - xv, f1 = xv;
    const float* cb = cores + (size_t)(m - 1) * (BOND * 2 * BOND);
    for (int i = t * 4; i < MAT_ELEMS; i += 256 * 4) {
        const int l = i >> 7;             // row (K)
        const int h = i & 127;            // col
        const float* rp = cb + l * 256 + h;
        const float4 a0 = *(const float4*)(rp);
        const float4 a1 = *(const float4*)(rp + BOND);
        const int base = (l >> 1) * 256 + (l & 1);
        dst[base + (h + 0) * 2] = f0 * a0.x + f1 * a1.x;
        dst[base + (h + 1) * 2] = f0 * a0.y + f1 * a1.y;
        dst[base + (h + 2) * 2] = f0 * a0.z + f1 * a1.z;
        dst[base + (h + 3) * 2] = f0 * a0.w + f1 * a1.w;
    }
}

__global__ void __launch_bounds__(256)
mps_level0_kernel(const float* __restrict__ x,
                  const float* __restrict__ core0,
                  const float* __restrict__ cores,
                  float* __restrict__ out,        // [b][128 nodes][16384]
                  float* __restrict__ lognorm) {  // [b*255 + node]
    __shared__ __align__(16) float sA[MAT_ELEMS];
    __shared__ __align__(16) float sB[MAT_ELEMS];
    __shared__ float sred[256];
    const int p = blockIdx.x;             // pair / output node 0..127
    const int b = blockIdx.y;             // batch 0..63
    build_site_rowmajor(sA, core0, cores, x, b, 2 * p);
    build_site_interleaved(sB, cores, x, b, 2 * p + 1);
    __syncthreads();
    mm128_norm_store(sA, sB,
                     out + ((size_t)b * 128 + p) * MAT_ELEMS,
                     lognorm + b * 255 + p, sred,
                     /*storeInterleaved=*/p & 1);
}

// ---------------------------------------------------------------------------
// Levels 1..7: contract pairs of previously normalized matrices (ping-pong).
// Node 2p is stored row-major, node 2p+1 interleaved, so staging is verbatim.
// ---------------------------------------------------------------------------
__global__ void __launch_bounds__(256)
mps_level_kernel(const float* __restrict__ in,
                 float* __restrict__ out,
                 float* __restrict__ lognorm,
                 int P, int logOff) {
    __shared__ __align__(16) float sA[MAT_ELEMS];
    __shared__ __align__(16) float sB[MAT_ELEMS];
    __shared__ float sred[256];
    const int p = blockIdx.x;
    const int b = blockIdx.y;
    const float* m1 = in + ((size_t)b * (2 * P) + 2 * p) * MAT_ELEMS;
    const float* m2 = m1 + MAT_ELEMS;
    for (int i = threadIdx.x * 4; i < MAT_ELEMS; i += 256 * 4) {
        *(float4*)(sA + i) = *(const float4*)(m1 + i);
        *(float4*)(sB + i) = *(const float4*)(m2 + i);
    }
    __syncthreads();
    mm128_norm_store(sA, sB,
                     out + ((size_t)b * P + p) * MAT_ELEMS,
                     lognorm + b * 255 + logOff + p, sred,
                     /*storeInterleaved=*/p & 1);
}

// ---------------------------------------------------------------------------
// Finalize: logits[b][o] = dot(row0(final), classifier[o]) + sum(log norms).
// Final matrix is node 0 -> stored row-major.
// ---------------------------------------------------------------------------
__global__ void __launch_bounds__(128)
mps_finalize_kernel(const float* __restrict__ lastBuf,   // [b][16384]
                    const float* __restrict__ lognorm,   // [b*255]
                    const float* __restrict__ classifier,// [10][128]
                    float* __restrict__ out) {           // [b][10]
    __shared__ float row[BOND];
    __shared__ float lsum;
    const int b = blockIdx.x;
    const int t = threadIdx.x;
    row[t] = lastBuf[(size_t)b * MAT_ELEMS + t];
    if (t == 0) {
        float s = 0.f;
        for (int i = 0; i < 255; ++i) s += lognorm[b * 255 + i];
        lsum = s;
    }
    __syncthreads();
    if (t < 10) {
        float acc = 0.f;
        for (int h = 0; h < BOND; ++h) acc += row[h] * classifier[t * BOND + h];
        out[b * 10 + t] = acc + lsum;
    }
}

// ---------------------------------------------------------------------------
// Launch.  ws layout:
//   bufA : 64*128*16384 f32 (512 MiB)  - even-level outputs
//   bufB : 64* 64*16384 f32 (256 MiB)  - odd-level outputs
//   logn : 64*255 f32                  - per-node log norms
// ---------------------------------------------------------------------------
extern "C" void kernel_launch(void* const* d_in, const int* in_sizes, int n_in,
                              void* d_out, int out_size, void* d_ws, size_t ws_size,
                              hipStream_t stream) {
    (void)in_sizes; (void)n_in; (void)out_size; (void)ws_size;
    const float* x          = (const float*)d_in[0];   // (64,256)
    const float* core0      = (const float*)d_in[1];   // (1,2,128)
    const float* cores      = (const float*)d_in[2];   // (255,128,2,128)
    const float* classifier = (const float*)d_in[3];   // (10,128)
    float* out = (float*)d_out;                        // (64,10)

    float* bufA = (float*)d_ws;
    float* bufB = bufA + (size_t)BATCH * 128 * MAT_ELEMS;
    float* logn = bufB + (size_t)BATCH * 64 * MAT_ELEMS;

    dim3 blk(256);

    // Level 0: 256 chain matrices -> 128 nodes.
    mps_level0_kernel<<<dim3(128, BATCH), blk, 0, stream>>>(x, core0, cores, bufA, logn);

    // Levels 1..7 ping-pong; log-norm offsets are cumulative node counts.
    struct { const char ab; int P; int off; } lv[7] = {
        {'B', 64, 128}, {'A', 32, 192}, {'B', 16, 224}, {'A', 8, 240},
        {'B', 4, 248}, {'A', 2, 252}, {'B', 1, 254}
    };
    const float* cur = bufA;
    for (int i = 0; i < 7; ++i) {
        float* dst = (lv[i].ab == 'A') ? bufA : bufB;
        mps_level_kernel<<<dim3(lv[i].P, BATCH), blk, 0, stream>>>(
            cur, dst, logn, lv[i].P, lv[i].off);
        cur = dst;
    }

    // Final matrix lives in bufB (level 7 output).
    mps_finalize_kernel<<<dim3(BATCH), dim3(128), 0, stream>>>(bufB, logn, classifier, out);
}